// AtomicPositionalEncoding_51994874085888
// MI455X (gfx1250) — compile-verified
//
#include <hip/hip_runtime.h>
#include <hip/hip_bf16.h>
#include <math.h>

typedef __attribute__((ext_vector_type(2))) float v2f;
typedef __attribute__((ext_vector_type(8))) float v8f;

#define D_MODEL 128
#define NUM_CLASSES 32
#define NPTS_PER_BATCH 1024
#define SQ_STRIDE 1028   // 1024 + 4 dword pad: k-rows map to distinct LDS banks

// Per-point feature computation shared by both passes.
// pr[k] = sh[k] * exp(-ETA*(1.5k - r)^2) * env(r), nan-cleaned; c = class id.
__device__ __forceinline__ void point_features(float px, float py, float pz, float pw,
                                               float pr[4], int& c, bool& valid)
{
    const float ETA = 4.0f;
    const float RC  = 6.0f;
    const float Y00 = 0.28209479177387814f;  // 0.5/sqrt(pi)
    const float C1  = 0.48860251190291992f;  // sqrt(3/(4pi))
    const float PI_ = 3.14159265358979323846f;

    float r2 = px * px + py * py + pz * pz;
    float r  = sqrtf(r2);
    valid = (r > 0.0f);
    float inv_r = valid ? (1.0f / r) : 1.0f;
    float ux = px * inv_r, uy = py * inv_r, uz = pz * inv_r;
    float rc  = fminf(r, RC);
    float env = 0.5f * cosf(PI_ * rc / RC) + 0.5f;
    float sh0 = Y00;
    float sh1 = C1 * uy;   // note reference order: y, z, x
    float sh2 = C1 * uz;
    float sh3 = C1 * ux;
    float sh[4] = { sh0, sh1, sh2, sh3 };
    #pragma unroll
    for (int k = 0; k < 4; ++k) {
        float d   = 1.5f * (float)k - r;
        float rad = expf(-ETA * d * d);
        float v   = sh[k] * rad * env;
        v = (v == v) ? v : 0.0f;             // jnp.nan_to_num(nan=0)
        pr[k] = v;
    }
    int ci = (int)pw;
    c = ci < 0 ? 0 : (ci > NUM_CLASSES - 1 ? NUM_CLASSES - 1 : ci);
}

// ---------------------------------------------------------------------------
// Kernel A: per-batch norm reduction over N via WMMA.
//   D[c,k] = sum_n onehot[n,c] * prod^2[n,k]   (a (32 x 1024) x (1024 x 4) GEMM)
// done as V_WMMA_F32_16X16X4_F32 tiles: M = class (two 16-wide tiles),
// N = k (4 of 16 used), K = 4 points per instruction.
// A-matrix 16x4 layout: V0 = (M=lane%16, K=2*(lane/16)), V1 = K+1.
// B-matrix 4x16 layout: V0 = (K=2*(lane/16), N=lane%16), V1 = K+1.
// D 16x16 layout: vgpr j -> (c=j, k=lane) lanes 0-15, (c=j+8, k=lane-16) lanes 16-31.
// Staging is transposed (s_sqT[k][n], padded stride) so each lane's two B
// elements are adjacent -> one unconditional ds_load_b64 per chunk, zeroing
// of the unused N columns done with v_cndmask instead of EXEC-masked loads.
// ---------------------------------------------------------------------------
__global__ void __launch_bounds__(256) pe_norm_kernel(const float* __restrict__ x,
                                                      float* __restrict__ inv_norm)
{
    __shared__ float s_sqT[4 * SQ_STRIDE];
    __shared__ int   s_cls[NPTS_PER_BATCH];
    __shared__ float s_acc[D_MODEL];

    const int b   = blockIdx.x;
    const int tid = threadIdx.x;
    if (tid < D_MODEL) s_acc[tid] = 0.0f;

    // Phase 1: stage sq[k][n] (transposed) and class into LDS.
    const float4* __restrict__ xin = (const float4*)x;
    #pragma unroll
    for (int i = 0; i < 4; ++i) {
        int n = tid + i * 256;
        float4 p = xin[b * NPTS_PER_BATCH + n];
        float pr[4]; int c; bool valid;
        point_features(p.x, p.y, p.z, p.w, pr, c, valid);
        #pragma unroll
        for (int k = 0; k < 4; ++k) {
            float v = valid ? pr[k] : 0.0f;
            s_sqT[k * SQ_STRIDE + n] = v * v;
        }
        s_cls[n] = valid ? c : -1;          // -1 never matches a class row
    }
    __syncthreads();

    // Phase 2: each wave reduces 128 points with 16x16x4 f32 WMMAs.
    const int wave = tid >> 5;
    const int lane = tid & 31;
    const int lpt  = lane & 15;   // M (class-in-tile) for A, N (k) for B
    const int hi   = lane >> 4;   // selects K pair {0,1} vs {2,3}
    const int krow = lpt & 3;     // clamped B row (masked below for lpt>=4)
    const bool bactive = (lpt < 4);

    v8f acc0 = {};                // classes 0..15
    v8f acc1 = {};                // classes 16..31
    const int base0 = wave * 128;
    for (int chunk = 0; chunk < 32; ++chunk) {
        int p0 = base0 + chunk * 4 + hi * 2;    // even -> 8B-aligned b64 loads

        // B pair: sq[p0][krow], sq[p0+1][krow] adjacent in transposed layout.
        const float2 braw = *reinterpret_cast<const float2*>(&s_sqT[krow * SQ_STRIDE + p0]);
        float b0v = bactive ? braw.x : 0.0f;
        float b1v = bactive ? braw.y : 0.0f;

        // Class pair: one b64.
        const int2 craw = *reinterpret_cast<const int2*>(&s_cls[p0]);
        int c0 = craw.x;
        int c1 = craw.y;

        v2f A0 = { (c0 == lpt)      ? 1.0f : 0.0f, (c1 == lpt)      ? 1.0f : 0.0f };
        v2f A1 = { (c0 == lpt + 16) ? 1.0f : 0.0f, (c1 == lpt + 16) ? 1.0f : 0.0f };
        v2f Bv = { b0v, b1v };
        acc0 = __builtin_amdgcn_wmma_f32_16x16x4_f32(
            false, A0, false, Bv, (short)0, acc0, false, false);
        acc1 = __builtin_amdgcn_wmma_f32_16x16x4_f32(
            false, A1, false, Bv, (short)0, acc1, false, false);
    }

    // Phase 3: cross-wave combine (only k<4 columns are meaningful).
    const int kk = lpt;
    if (kk < 4) {
        #pragma unroll
        for (int j = 0; j < 8; ++j) {
            int c = j + hi * 8;
            atomicAdd(&s_acc[kk * NUM_CLASSES + c],      acc0[j]);
            atomicAdd(&s_acc[kk * NUM_CLASSES + c + 16], acc1[j]);
        }
    }
    __syncthreads();

    if (tid < D_MODEL) {
        float s   = s_acc[tid];
        float nrm = sqrtf(s);
        inv_norm[b * D_MODEL + tid] = 1.0f / fmaxf(nrm, 1e-12f);
    }
}

// ---------------------------------------------------------------------------
// Kernel B: one wave per point. Recompute features, apply inverse norm,
// closed-form mean/std (ddof=1) from the 4 nonzeros, stream out 128 floats
// as one float4 per lane (512B contiguous per wave).
// ---------------------------------------------------------------------------
__global__ void __launch_bounds__(256) pe_out_kernel(const float* __restrict__ x,
                                                     const float* __restrict__ inv_norm,
                                                     float* __restrict__ out,
                                                     int npts)
{
    int pt   = blockIdx.x * 8 + (threadIdx.x >> 5);
    int lane = threadIdx.x & 31;
    if (pt >= npts) return;
    int b = pt >> 10;                     // N = 1024 points per batch

    const float4 p = ((const float4*)x)[pt];
    float pr[4]; int c; bool valid;
    point_features(p.x, p.y, p.z, p.w, pr, c, valid);

    const float* nb = inv_norm + b * D_MODEL + c;
    float v0 = (valid ? pr[0] : 0.0f) * nb[0];
    float v1 = (valid ? pr[1] : 0.0f) * nb[32];
    float v2 = (valid ? pr[2] : 0.0f) * nb[64];
    float v3 = (valid ? pr[3] : 0.0f) * nb[96];

    float sum   = v0 + v1 + v2 + v3;
    float sumsq = v0 * v0 + v1 * v1 + v2 * v2 + v3 * v3;
    float mean  = sum * (1.0f / 128.0f);
    float var   = (sumsq - 128.0f * mean * mean) * (1.0f / 127.0f);
    var = fmaxf(var, 0.0f);
    float s  = 1.0f / (sqrtf(var) + 1e-6f);
    float bg = -mean * s;                 // value of the 124 zero features

    int kblk  = lane >> 3;                // which radial block this lane's d's fall in
    int posin = (lane * 4) & 31;          // position within the 32-wide class block
    float vsel = (kblk == 0) ? v0 : (kblk == 1) ? v1 : (kblk == 2) ? v2 : v3;
    float sp   = (vsel - mean) * s;

    float4 o;
    o.x = (posin + 0 == c) ? sp : bg;
    o.y = (posin + 1 == c) ? sp : bg;
    o.z = (posin + 2 == c) ? sp : bg;
    o.w = (posin + 3 == c) ? sp : bg;
    ((float4*)out)[(size_t)pt * 32 + lane] = o;
}

extern "C" void kernel_launch(void* const* d_in, const int* in_sizes, int n_in,
                              void* d_out, int out_size, void* d_ws, size_t ws_size,
                              hipStream_t stream)
{
    const float* x = (const float*)d_in[0];
    float* inv_norm = (float*)d_ws;               // 256*128*4 = 128 KB scratch
    int npts   = in_sizes[0] / 4;                 // B*N = 262144
    int nbatch = npts / NPTS_PER_BATCH;           // 256

    pe_norm_kernel<<<nbatch, 256, 0, stream>>>(x, inv_norm);
    pe_out_kernel<<<npts / 8, 256, 0, stream>>>(x, inv_norm, (float*)d_out, npts);
}